// GCN_71751723647375
// MI455X (gfx1250) — compile-verified
//
#include <hip/hip_runtime.h>
#include <hip/hip_bf16.h>
#include <stdint.h>
#include <math.h>

#define N_NODES 100000
#define N_EDGES 3200000
#define C_IN    128
#define HIDDEN  64
#define C_OUT   40

typedef __attribute__((ext_vector_type(2))) float v2f;
typedef __attribute__((ext_vector_type(8))) float v8f;

// ---------------------------------------------------------------- zero fill
__global__ void gcn_zero_kernel(float* __restrict__ p, long n) {
    long i = (long)blockIdx.x * blockDim.x + threadIdx.x;
    long stride = (long)gridDim.x * blockDim.x;
    for (; i < n; i += stride) p[i] = 0.0f;
}

// ---------------------------------------------------------------- in-degree
__global__ void gcn_degree_kernel(const long long* __restrict__ ei,
                                  float* __restrict__ deg, int nE) {
    int e = blockIdx.x * blockDim.x + threadIdx.x;
    if (e < nE) {
        int dst = (int)ei[(long)nE + e];
        atomicAdd(&deg[dst], 1.0f);
    }
}

// ---------------------------------------------------------------- deg -> deg^{-1/2}
__global__ void gcn_dinv_kernel(float* __restrict__ deg, int n) {
    int i = blockIdx.x * blockDim.x + threadIdx.x;
    if (i < n) {
        float d = deg[i];
        deg[i] = (d > 0.0f) ? rsqrtf(d) : 0.0f;
    }
}

// ---------------------------------------------------------------- WMMA f32 GEMM
// C[M,N] = A[M,K] * B[K,N]; row-major; M % 16 == 0, K % 4 == 0; N guarded.
// One wave computes one 16x16 tile of C via V_WMMA_F32_16X16X4_F32.
// Out-of-range columns: clamp the B address (garbage only lands in that lane's
// own output column, which is never stored) -> branch-free inner loop.
__global__ void gcn_wmma_gemm_f32(const float* __restrict__ A,
                                  const float* __restrict__ B,
                                  float* __restrict__ C,
                                  int M, int K, int N, int nTilesN) {
    int wavesPerBlock = blockDim.x >> 5;
    int waveId = blockIdx.x * wavesPerBlock + (threadIdx.x >> 5);
    int lane   = threadIdx.x & 31;
    int tileM  = waveId / nTilesN;
    int tileN  = waveId - tileM * nTilesN;
    if (tileM * 16 >= M) return;          // wave-uniform guard: EXEC stays all-1s

    int row  = tileM * 16 + (lane & 15);  // A row for this lane
    int col  = tileN * 16 + (lane & 15);  // B/C column for this lane
    int koff = (lane >> 4) << 1;          // lanes 0-15 -> K+0,K+1 ; lanes 16-31 -> K+2,K+3
    bool colOK = (col < N);
    int colSafe = colOK ? col : 0;        // clamped, loop-invariant

    const float* __restrict__ arow = A + (long)row * K + koff;
    const float* __restrict__ bcol = B + (long)koff * N + colSafe;

    v8f acc = {};
    for (int k0 = 0; k0 < K; k0 += 4) {
        v2f a;
        a.x = arow[0];
        a.y = arow[1];
        v2f b;
        b.x = bcol[0];
        b.y = bcol[N];
        acc = __builtin_amdgcn_wmma_f32_16x16x4_f32(
            /*neg_a=*/false, a, /*neg_b=*/false, b,
            /*c_mod=*/(short)0, acc, /*reuse_a=*/false, /*reuse_b=*/false);
        arow += 4;
        bcol += (long)4 * N;
    }

    // D layout: VGPR r -> lanes 0-15: M=r ; lanes 16-31: M=8+r ; N = lane&15
    int rbase = tileM * 16 + ((lane >> 4) << 3);
    if (colOK) {
        #pragma unroll
        for (int r = 0; r < 8; ++r)
            C[(long)(rbase + r) * N + col] = acc[r];
    }
}

// ---------------------------------------------------------------- edge scatter-add
// chunksPerEdge lanes per edge, each lane moves 4 features (float4 gather + 4 atomics)
__global__ void gcn_scatter_kernel(const long long* __restrict__ ei,
                                   const float* __restrict__ dinv,
                                   const float* __restrict__ feat,
                                   float* __restrict__ out,
                                   int nE, int F, int chunksPerEdge) {
    long t = (long)blockIdx.x * blockDim.x + threadIdx.x;
    long total = (long)nE * chunksPerEdge;
    if (t >= total) return;
    int e = (int)(t / chunksPerEdge);
    int q = (int)(t - (long)e * chunksPerEdge);
    int src = (int)ei[e];
    int dst = (int)ei[(long)nE + e];
    float norm = dinv[src] * dinv[dst];
    const float4 v = *(const float4*)(feat + (long)src * F + q * 4);
    float* o = out + (long)dst * F + q * 4;
    atomicAdd(o + 0, v.x * norm);
    atomicAdd(o + 1, v.y * norm);
    atomicAdd(o + 2, v.z * norm);
    atomicAdd(o + 3, v.w * norm);
}

// ---------------------------------------------------------------- bias + ReLU (in place)
__global__ void gcn_relu_bias_kernel(float* __restrict__ h,
                                     const float* __restrict__ b,
                                     long total, int F) {
    long i = (long)blockIdx.x * blockDim.x + threadIdx.x;
    if (i < total) {
        int c = (int)(i % F);
        float v = h[i] + b[c];
        h[i] = v > 0.0f ? v : 0.0f;
    }
}

// ---------------------------------------------------------------- bias + log_softmax (in place)
__global__ void gcn_logsoftmax_kernel(float* __restrict__ out,
                                      const float* __restrict__ b,
                                      int n, int F) {
    int i = blockIdx.x * blockDim.x + threadIdx.x;
    if (i >= n) return;
    float* row = out + (long)i * F;
    float mx = -INFINITY;
    for (int c = 0; c < F; ++c) {
        float v = row[c] + b[c];
        row[c] = v;
        mx = fmaxf(mx, v);
    }
    float s = 0.0f;
    for (int c = 0; c < F; ++c) s += expf(row[c] - mx);
    float lse = mx + logf(s);
    for (int c = 0; c < F; ++c) row[c] -= lse;
}

// ---------------------------------------------------------------- launch
extern "C" void kernel_launch(void* const* d_in, const int* in_sizes, int n_in,
                              void* d_out, int out_size, void* d_ws, size_t ws_size,
                              hipStream_t stream) {
    const float*     x  = (const float*)d_in[0];
    const long long* ei = (const long long*)d_in[1];   // int64 [2, E]
    const float*     W1 = (const float*)d_in[2];       // [128, 64]
    const float*     b1 = (const float*)d_in[3];       // [64]
    const float*     W2 = (const float*)d_in[4];       // [64, 40]
    const float*     b2 = (const float*)d_in[5];       // [40]
    float* out = (float*)d_out;                        // [N, 40]

    // workspace layout (floats): deg(N) | agg1(N*64) | xW1(N*64) | hW2(N*40)
    float* ws   = (float*)d_ws;
    float* deg  = ws;                                  // -> becomes dinv in place
    float* agg1 = deg  + N_NODES;
    float* xW1  = agg1 + (long)N_NODES * HIDDEN;
    float* hW2  = xW1  + (long)N_NODES * HIDDEN;

    const int B = 256;

    // 0) zero accumulation targets: deg+agg1 (contiguous) and d_out
    {
        long n0 = (long)N_NODES * (1 + HIDDEN);
        gcn_zero_kernel<<<4096, B, 0, stream>>>(ws, n0);
        long n1 = (long)N_NODES * C_OUT;
        gcn_zero_kernel<<<4096, B, 0, stream>>>(out, n1);
    }

    // 1) in-degree, then dinv = deg^{-1/2}
    gcn_degree_kernel<<<(N_EDGES + B - 1) / B, B, 0, stream>>>(ei, deg, N_EDGES);
    gcn_dinv_kernel<<<(N_NODES + B - 1) / B, B, 0, stream>>>(deg, N_NODES);

    // 2) xW1 = x @ W1   (M=100000, K=128, N=64; 6250 x 4 tiles)
    {
        int tilesM = N_NODES / 16, tilesN = HIDDEN / 16;
        int waves = tilesM * tilesN;
        int blocks = (waves + (B / 32) - 1) / (B / 32);
        gcn_wmma_gemm_f32<<<blocks, B, 0, stream>>>(x, W1, xW1, N_NODES, C_IN, HIDDEN, tilesN);
    }

    // 3) agg1 = scatter_add(norm * xW1[src] -> dst)   (16 lanes/edge * 4 feats)
    {
        long total = (long)N_EDGES * (HIDDEN / 4);
        int blocks = (int)((total + B - 1) / B);
        gcn_scatter_kernel<<<blocks, B, 0, stream>>>(ei, deg, xW1, agg1, N_EDGES, HIDDEN, HIDDEN / 4);
    }

    // 4) h = relu(agg1 + b1), in place
    {
        long total = (long)N_NODES * HIDDEN;
        int blocks = (int)((total + B - 1) / B);
        gcn_relu_bias_kernel<<<blocks, B, 0, stream>>>(agg1, b1, total, HIDDEN);
    }

    // 5) hW2 = h @ W2   (M=100000, K=64, N=40; 6250 x 3 tiles, col-guarded)
    {
        int tilesM = N_NODES / 16, tilesN = (C_OUT + 15) / 16;
        int waves = tilesM * tilesN;
        int blocks = (waves + (B / 32) - 1) / (B / 32);
        gcn_wmma_gemm_f32<<<blocks, B, 0, stream>>>(agg1, W2, hW2, N_NODES, HIDDEN, C_OUT, tilesN);
    }

    // 6) out = scatter_add(norm * hW2[src] -> dst)   (10 lanes/edge * 4 feats)
    {
        long total = (long)N_EDGES * (C_OUT / 4);
        int blocks = (int)((total + B - 1) / B);
        gcn_scatter_kernel<<<blocks, B, 0, stream>>>(ei, deg, hW2, out, N_EDGES, C_OUT, C_OUT / 4);
    }

    // 7) out = log_softmax(out + b2), in place
    gcn_logsoftmax_kernel<<<(N_NODES + B - 1) / B, B, 0, stream>>>(out, b2, N_NODES, C_OUT);
}